// TMoELayer_72859825209500
// MI455X (gfx1250) — compile-verified
//
#include <hip/hip_runtime.h>
#include <math.h>

typedef __attribute__((ext_vector_type(2))) float v2f;
typedef __attribute__((ext_vector_type(8))) float v8f;

#define D_IN   1024
#define D_OUT  1024
#define NEXP   8
#define RANK   16
#define KDIM   128          // NEXP * RANK
#define SCALING 2.0f        // alpha/r = 32/16

// ---------------------------------------------------------------------------
// Kernel 0: repack routed [E, D_OUT, R] -> W2 [K=E*R, D_OUT] row-major
// ---------------------------------------------------------------------------
__global__ __launch_bounds__(256) void repack_w2(const float* __restrict__ routed,
                                                 float* __restrict__ w2) {
    int idx = blockIdx.x * 256 + threadIdx.x;   // 0 .. 128*1024-1
    int k = idx >> 10;                          // 0..127
    int o = idx & 1023;
    int e = k >> 4;
    int r = k & 15;
    w2[idx] = routed[(e * D_OUT + o) * RANK + r];
}

// ---------------------------------------------------------------------------
// Kernel 1: fused router + compress + gating.
// One wave32 handles 4 tokens. Each lane accumulates float4 partials of all
// 24 dot products (8 router + 16 compress), butterfly-reduces so every lane
// holds every sum, then writes the scaled gate*ce outer product row (128 f32).
// ---------------------------------------------------------------------------
__global__ __launch_bounds__(256) void route_compress(const float* __restrict__ x,
                                                      const float* __restrict__ w_route,
                                                      const float* __restrict__ compress,
                                                      float* __restrict__ tmp) {
    const int lane = threadIdx.x & 31;
    const int wave = threadIdx.x >> 5;
    const int tok0 = (blockIdx.x * 8 + wave) * 4;

    float acc[24][4];
#pragma unroll
    for (int j = 0; j < 24; ++j)
#pragma unroll
        for (int t = 0; t < 4; ++t) acc[j][t] = 0.0f;

    // 1024 / (32 lanes * 4 floats) = 8 strips
    for (int it = 0; it < 8; ++it) {
        const int off = it * 128 + lane * 4;
        float4 xv[4];
#pragma unroll
        for (int t = 0; t < 4; ++t)
            xv[t] = *(const float4*)(x + (size_t)(tok0 + t) * D_IN + off);
#pragma unroll
        for (int j = 0; j < 8; ++j) {
            float4 w = *(const float4*)(w_route + j * D_IN + off);
#pragma unroll
            for (int t = 0; t < 4; ++t)
                acc[j][t] += w.x * xv[t].x + w.y * xv[t].y + w.z * xv[t].z + w.w * xv[t].w;
        }
#pragma unroll
        for (int j = 0; j < 16; ++j) {
            float4 w = *(const float4*)(compress + j * D_IN + off);
#pragma unroll
            for (int t = 0; t < 4; ++t)
                acc[8 + j][t] += w.x * xv[t].x + w.y * xv[t].y + w.z * xv[t].z + w.w * xv[t].w;
        }
    }

    // butterfly reduce across wave32: every lane ends with the full sums
#pragma unroll
    for (int j = 0; j < 24; ++j)
#pragma unroll
        for (int t = 0; t < 4; ++t) {
            float v = acc[j][t];
#pragma unroll
            for (int s = 16; s >= 1; s >>= 1) v += __shfl_xor(v, s, 32);
            acc[j][t] = v;
        }

#pragma unroll
    for (int t = 0; t < 4; ++t) {
        // top-2 over the 8 router logits (first-index wins ties, like lax.top_k)
        float v0 = acc[0][t]; int i0 = 0;
#pragma unroll
        for (int j = 1; j < 8; ++j) { float l = acc[j][t]; if (l > v0) { v0 = l; i0 = j; } }
        float v1 = -3.4e38f; int i1 = 0;
#pragma unroll
        for (int j = 0; j < 8; ++j) {
            if (j == i0) continue;
            float l = acc[j][t]; if (l > v1) { v1 = l; i1 = j; }
        }
        // softmax over [v0, v1] (v0 >= v1)
        float ex = __expf(v1 - v0);
        float g0 = 1.0f / (1.0f + ex);
        float g1 = ex   / (1.0f + ex);

        // this lane writes k = lane*4 .. lane*4+3  (all within one expert block)
        const int kbase = lane * 4;
        const int e     = lane >> 2;            // kbase >> 4
        const int sel   = lane & 3;             // which ce quadruple: r = sel*4 .. sel*4+3
        float g = (e == i0) ? g0 : ((e == i1) ? g1 : 0.0f);
        g *= SCALING;

        // compile-time-indexed select of ce[sel*4 + 0..3] (avoids reg-array spill)
        float c0 = 0.f, c1 = 0.f, c2 = 0.f, c3 = 0.f;
#pragma unroll
        for (int s = 0; s < 4; ++s) {
            bool p = (sel == s);
            c0 = p ? acc[8 + s * 4 + 0][t] : c0;
            c1 = p ? acc[8 + s * 4 + 1][t] : c1;
            c2 = p ? acc[8 + s * 4 + 2][t] : c2;
            c3 = p ? acc[8 + s * 4 + 3][t] : c3;
        }
        float4 outv = make_float4(g * c0, g * c1, g * c2, g * c3);
        *(float4*)(tmp + (size_t)(tok0 + t) * KDIM + kbase) = outv;
    }
}

// ---------------------------------------------------------------------------
// Kernel 2: out[NTOK,1024] = tmp[NTOK,128] @ W2[128,1024] via fp32 WMMA.
// One wave per 16x16 C tile, 32 chained V_WMMA_F32_16X16X4_F32 over K=128.
//   A 16x4 layout : lane%16 = M, K = (lane/16)*2 + v  -> consecutive float2
//   B 4x16 layout : lane%16 = N, K = (lane/16)*2 + v
//   C 16x16 layout: lane%16 = N, M = (lane/16)*8 + v
// ---------------------------------------------------------------------------
__global__ __launch_bounds__(256) void gemm_wmma(const float* __restrict__ A,
                                                 const float* __restrict__ B,
                                                 float* __restrict__ C) {
    const int lane = threadIdx.x & 31;
    const int wave = threadIdx.x >> 5;
    const int tile = blockIdx.x * 8 + wave;
    const int tN   = tile & 63;                 // 1024/16 = 64 N tiles
    const int tM   = tile >> 6;
    const int half = lane >> 4;                 // 0 or 1
    const int l16  = lane & 15;

    const float* aRow = A + (size_t)(tM * 16 + l16) * KDIM + half * 2;
    const float* bCol = B + (size_t)(half * 2) * D_OUT + tN * 16 + l16;

    v8f c = {0.f, 0.f, 0.f, 0.f, 0.f, 0.f, 0.f, 0.f};
#pragma unroll
    for (int ks = 0; ks < 32; ++ks) {
        v2f a = *(const v2f*)(aRow + ks * 4);
        v2f b;
        b[0] = bCol[(size_t)(ks * 4 + 0) * D_OUT];
        b[1] = bCol[(size_t)(ks * 4 + 1) * D_OUT];
        c = __builtin_amdgcn_wmma_f32_16x16x4_f32(false, a, false, b,
                                                  (short)0, c, false, false);
    }

    float* outp = C + (size_t)(tM * 16 + half * 8) * D_OUT + tN * 16 + l16;
#pragma unroll
    for (int v = 0; v < 8; ++v) outp[(size_t)v * D_OUT] = c[v];
}

// ---------------------------------------------------------------------------
extern "C" void kernel_launch(void* const* d_in, const int* in_sizes, int n_in,
                              void* d_out, int out_size, void* d_ws, size_t ws_size,
                              hipStream_t stream) {
    const float* x        = (const float*)d_in[0];
    const float* w_route  = (const float*)d_in[1];
    const float* compress = (const float*)d_in[2];
    const float* routed   = (const float*)d_in[3];
    float*       out      = (float*)d_out;

    float* w2  = (float*)d_ws;                  // 128*1024 floats = 512 KB
    float* tmp = w2 + KDIM * D_OUT;             // ntok*128 floats = 8 MB

    const int ntok = in_sizes[0] / D_IN;        // 16384

    // W2 repack (independent of tmp build; stream order gives gemm both)
    repack_w2<<<(KDIM * D_OUT) / 256, 256, 0, stream>>>(routed, w2);

    // fused router + compress + gate  (32 tokens per 256-thread block)
    route_compress<<<ntok / 32, 256, 0, stream>>>(x, w_route, compress, tmp);

    // WMMA combine GEMM: (ntok/16)*(1024/16) tiles, 8 waves per block
    const int tiles = (ntok / 16) * (D_OUT / 16);
    gemm_wmma<<<tiles / 8, 256, 0, stream>>>(tmp, w2, out);
}